// PredictYolov3_33208687132985
// MI455X (gfx1250) — compile-verified
//
#include <hip/hip_runtime.h>
#include <stdint.h>

// ---- problem constants (from reference) ----
#define B_    64          // batch
#define A_    22743       // anchors: 3*(19^2 + 38^2 + 76^2)
#define APAD_ 22752       // padded to 711 chunks of 32
#define NCH_  711         // number of 32-wide leaf chunks
#define CH_   85          // 4 box + 1 obj + 80 classes
#define NC_   80
#define K_    300
#define NW_   10          // (K_+31)/32 keep-mask words
#define L0_   1083        // 3*19*19
#define L1_   5415        // + 3*38*38
#define THR_NMS_ 0.3f
#define CLS_OFF_ 1000.0f

typedef unsigned int u32x4 __attribute__((ext_vector_type(4)));
typedef int          i32x8 __attribute__((ext_vector_type(8)));
typedef int          i32x4 __attribute__((ext_vector_type(4)));

// -------------------------------------------------------------------------
// Pass 1: objectness score gather.  score = sigmoid(x) if x>0 else 0
// (sigmoid(x) > 0.5  <=>  x > 0).  Only channel 4 of each 340B anchor row
// is touched -> ~1 cacheline per anchor instead of full payload.
// -------------------------------------------------------------------------
__global__ void yolo_score_pass(const float* __restrict__ p, float* __restrict__ sc)
{
    int i = blockIdx.x * blockDim.x + threadIdx.x;
    if (i >= B_ * A_) return;
    float lg = p[(size_t)i * CH_ + 4];
    sc[i] = (lg > 0.0f) ? (1.0f / (1.0f + __expf(-lg))) : 0.0f;
}

// -------------------------------------------------------------------------
// Pass 2: one workgroup (1024 thr = 32 waves) per image.
//   a) TDM-DMA the score row into LDS (tensor_load_to_lds, TENSORcnt)
//   b) tournament top-K=300: chunk-max level + single-wave selection
//   c) decode + class argmax for the 300 survivors only
//   d) NMS: parallel suppression bitmask, then single-wave bit sweep
//   e) write padded outputs (ids | boxes | labels | scores) as f32
// -------------------------------------------------------------------------
extern __shared__ float s_dyn[];   // APAD_ floats of dynamic LDS (~91 KB)

__global__ __launch_bounds__(1024, 1)
void yolo_topk_nms(const float* __restrict__ p, const float* __restrict__ ancs,
                   const float* __restrict__ sc, float* __restrict__ out)
{
    const int b   = blockIdx.x;
    const int tid = threadIdx.x;
    float* s_sc = s_dyn;                 // [APAD_] scores (consumed destructively)

    __shared__ int      sel_idx[K_];
    __shared__ float    sel_val[K_];
    __shared__ float    s_ob[K_ * 4];    // class-offset boxes (for IoU)
    __shared__ float    s_bx[K_ * 4];    // raw ltrb boxes (for output)
    __shared__ float    s_area[K_];
    __shared__ int      s_lab[K_];
    __shared__ int      s_valid[K_];
    __shared__ unsigned s_mask[K_ * NW_]; // suppression relation bitmask
    __shared__ unsigned s_kw[NW_];        // final keep words
    __shared__ float    lvl_v[NCH_];      // per-chunk max value
    __shared__ int      lvl_i[NCH_];      // per-chunk argmax (global idx)

    const float* src = sc + (size_t)b * A_;

    // ---- (a) stage score row into LDS via the Tensor Data Mover ----
#if defined(__gfx1250__) && __has_builtin(__builtin_amdgcn_tensor_load_to_lds)
    if (tid < 32) {   // one wave issues the single DMA descriptor
        uint64_t ga  = (uint64_t)(uintptr_t)src;
        uint32_t lds = (uint32_t)(uintptr_t)s_sc;     // LDS aperture: low 32 bits
        u32x4 g0;
        g0[0] = 1u;                                   // count=1, user descriptor
        g0[1] = lds;                                  // lds_addr
        g0[2] = (uint32_t)ga;                         // global_addr[31:0]
        g0[3] = (uint32_t)((ga >> 32) & 0x01ffffffu)  // global_addr[56:32]
              | (2u << 30);                           // type = 2 ("image")
        const unsigned td0 = A_, td1 = 1u, tl0 = A_, tl1 = 1u, tl2 = 0u;
        const unsigned long long st0 = A_, st1 = A_;
        i32x8 g1;
        g1[0] = (int)(2u << 16);                                  // data_size = 4B
        g1[1] = (int)((td0 & 0xffffu) << 16);                     // tensor_dim0 lo
        g1[2] = (int)((td0 >> 16) | ((td1 & 0xffffu) << 16));     // dim0 hi | dim1 lo
        g1[3] = (int)((td1 >> 16) | (tl0 << 16));                 // dim1 hi | tile_dim0
        g1[4] = (int)(tl1 | (tl2 << 16));                         // tile_dim1 | tile_dim2
        g1[5] = (int)(unsigned)(st0 & 0xffffffffu);               // dim0_stride lo32
        g1[6] = (int)((unsigned)(st0 >> 32) | ((unsigned)(st1 & 0xffffu) << 16));
        g1[7] = (int)(unsigned)(st1 >> 16);                       // dim1_stride[47:16]
        i32x4 z4 = (i32x4){0, 0, 0, 0};                           // 2-D: groups 2/3 unused
        i32x8 z8 = (i32x8){0, 0, 0, 0, 0, 0, 0, 0};
        __builtin_amdgcn_tensor_load_to_lds(g0, g1, z4, z4, z8, 0);
        __builtin_amdgcn_s_wait_tensorcnt(0);
    }
    if (tid >= 32 && tid < 32 + (APAD_ - A_)) s_sc[A_ + (tid - 32)] = -2.0f; // pad tail
    __syncthreads();
#else
    for (int i = tid; i < A_; i += 1024) s_sc[i] = src[i];
    if (tid < (APAD_ - A_)) s_sc[A_ + tid] = -2.0f;
    __syncthreads();
#endif

    // ---- (b1) build the chunk-max level: one wave per chunk, strided ----
    {
        const int w = tid >> 5, lane = tid & 31;
        for (int c = w; c < NCH_; c += 32) {
            float bv = s_sc[c * 32 + lane];
            int   bi = c * 32 + lane;
            #pragma unroll
            for (int m = 16; m > 0; m >>= 1) {
                float ov = __shfl_xor(bv, m);
                int   oi = __shfl_xor(bi, m);
                if (ov > bv || (ov == bv && oi < bi)) { bv = ov; bi = oi; }
            }
            if (lane == 0) { lvl_v[c] = bv; lvl_i[c] = bi; }
        }
    }
    __syncthreads();

    // ---- (b2) 300 selection steps on a single wave ----
    if (tid < 32) {
        const int lane = tid;
        for (int k = 0; k < K_; ++k) {
            // scan the 711-entry level (23 per lane, conflict-free)
            float bv = -3.0f; int bi = 0x7fffffff;
            for (int c = lane; c < NCH_; c += 32) {
                float v = lvl_v[c]; int i = lvl_i[c];
                if (v > bv || (v == bv && i < bi)) { bv = v; bi = i; }
            }
            #pragma unroll
            for (int m = 16; m > 0; m >>= 1) {
                float ov = __shfl_xor(bv, m);
                int   oi = __shfl_xor(bi, m);
                if (ov > bv || (ov == bv && oi < bi)) { bv = ov; bi = oi; }
            }
            // all lanes now agree on winner (bv, bi)
            if (lane == 0) { sel_idx[k] = bi; sel_val[k] = bv; }
            const int c0 = bi >> 5;
            // rebuild chunk c0 with the winner removed
            float v = s_sc[c0 * 32 + lane];
            if (lane == (bi & 31)) { v = -2.0f; s_sc[bi] = -2.0f; }
            int i = c0 * 32 + lane;
            #pragma unroll
            for (int m = 16; m > 0; m >>= 1) {
                float ov = __shfl_xor(v, m);
                int   oi = __shfl_xor(i, m);
                if (ov > v || (ov == v && oi < i)) { v = ov; i = oi; }
            }
            if (lane == 0) { lvl_v[c0] = v; lvl_i[c0] = i; }
        }
    }
    __syncthreads();

    // ---- (c) decode boxes + class argmax for the 300 survivors ----
    if (tid < K_) {
        const int   a  = sel_idx[tid];
        const float sv = sel_val[tid];
        const float* pr = p + ((size_t)b * A_ + a) * CH_;
        float tx = pr[0], ty = pr[1], tw = pr[2], th = pr[3];
        float bc = pr[5]; int bl = 0;
        #pragma unroll 4
        for (int c = 1; c < NC_; ++c) {      // argmax over logits == argmax sigmoid
            float v = pr[5 + c];
            if (v > bc) { bc = v; bl = c; }
        }
        const int label = bl + 1;
        const float fs = (a < L0_) ? 19.0f : ((a < L1_) ? 38.0f : 76.0f);
        const float ax = ancs[a * 4 + 0], ay = ancs[a * 4 + 1];
        const float aw = ancs[a * 4 + 2], ah = ancs[a * 4 + 3];
        const float cx = tx / fs + ax, cy = ty / fs + ay;
        const float w  = __expf(tw) * aw, h = __expf(th) * ah;
        const float l = cx - 0.5f * w, t = cy - 0.5f * h;
        const float r = cx + 0.5f * w, bt = cy + 0.5f * h;
        s_bx[tid * 4 + 0] = l;  s_bx[tid * 4 + 1] = t;
        s_bx[tid * 4 + 2] = r;  s_bx[tid * 4 + 3] = bt;
        const float off = (float)label * CLS_OFF_;
        s_ob[tid * 4 + 0] = l + off;  s_ob[tid * 4 + 1] = t + off;
        s_ob[tid * 4 + 2] = r + off;  s_ob[tid * 4 + 3] = bt + off;
        s_area[tid] = fmaxf(r - l, 0.0f) * fmaxf(bt - t, 0.0f);
        s_lab[tid]  = label;
        s_valid[tid] = (sv > 0.0f) ? 1 : 0;
    }
    __syncthreads();

    // ---- (d1) parallel suppression relation:
    //      s_mask[i][w] bit j' set  <=>  j = w*32+j' > i  &&  iou(i,j) > thr
    for (int t = tid; t < K_ * NW_; t += 1024) {
        const int i = t / NW_, w = t % NW_;
        const float il = s_ob[i * 4 + 0], it = s_ob[i * 4 + 1];
        const float ir = s_ob[i * 4 + 2], ib = s_ob[i * 4 + 3];
        const float ia = s_area[i];
        unsigned m = 0u;
        #pragma unroll 4
        for (int jp = 0; jp < 32; ++jp) {
            const int j = w * 32 + jp;
            if (j < K_ && j > i) {
                const float xl = fmaxf(il, s_ob[j * 4 + 0]);
                const float yt = fmaxf(it, s_ob[j * 4 + 1]);
                const float xr = fminf(ir, s_ob[j * 4 + 2]);
                const float yb = fminf(ib, s_ob[j * 4 + 3]);
                const float inter = fmaxf(xr - xl, 0.0f) * fmaxf(yb - yt, 0.0f);
                const float iou = inter / (ia + s_area[j] - inter + 1e-9f);
                if (iou > THR_NMS_) m |= (1u << jp);
            }
        }
        s_mask[i * NW_ + w] = m;
    }
    // initial keep words from validity
    if (tid < NW_) {
        unsigned kw = 0u;
        for (int jp = 0; jp < 32; ++jp) {
            const int j = tid * 32 + jp;
            if (j < K_ && s_valid[j]) kw |= (1u << jp);
        }
        s_kw[tid] = kw;
    }
    __syncthreads();

    // ---- (d2) sequential suppression sweep on a single wave (no barriers) ----
    if (tid < 32) {
        const int lane = tid;
        unsigned kw = (lane < NW_) ? s_kw[lane] : 0u;
        for (int i = 0; i < K_; ++i) {
            const unsigned owner_kw = __shfl(kw, i >> 5);
            const int ki = (owner_kw >> (i & 31)) & 1;
            if (ki && lane < NW_) kw &= ~s_mask[i * NW_ + lane];
        }
        if (lane < NW_) s_kw[lane] = kw;
    }
    __syncthreads();

    // ---- (e) outputs: ids | boxes | labels | scores (flat, return order) ----
    if (tid < K_) {
        const int    kp   = (s_kw[tid >> 5] >> (tid & 31)) & 1;
        const size_t base = (size_t)b * K_ + tid;
        float* o_ids = out;
        float* o_box = out + (size_t)B_ * K_;
        float* o_lab = out + (size_t)B_ * K_ * 5;
        float* o_sco = out + (size_t)B_ * K_ * 6;
        o_ids[base] = kp ? (float)b : -1.0f;
        o_box[base * 4 + 0] = kp ? s_bx[tid * 4 + 0] : 0.0f;
        o_box[base * 4 + 1] = kp ? s_bx[tid * 4 + 1] : 0.0f;
        o_box[base * 4 + 2] = kp ? s_bx[tid * 4 + 2] : 0.0f;
        o_box[base * 4 + 3] = kp ? s_bx[tid * 4 + 3] : 0.0f;
        o_lab[base] = kp ? (float)s_lab[tid] : 0.0f;
        o_sco[base] = kp ? sel_val[tid] : 0.0f;
    }
}

extern "C" void kernel_launch(void* const* d_in, const int* in_sizes, int n_in,
                              void* d_out, int out_size, void* d_ws, size_t ws_size,
                              hipStream_t stream)
{
    const float* p    = (const float*)d_in[0];   // [B, A, 85]
    const float* ancs = (const float*)d_in[1];   // [A, 4]
    float* scores = (float*)d_ws;                // B*A floats of scratch
    float* out    = (float*)d_out;

    const int total = B_ * A_;
    yolo_score_pass<<<(total + 255) / 256, 256, 0, stream>>>(p, scores);

    const size_t smem = (size_t)APAD_ * sizeof(float);   // ~91 KB dynamic LDS
    (void)hipFuncSetAttribute((const void*)yolo_topk_nms,
                              hipFuncAttributeMaxDynamicSharedMemorySize,
                              (int)smem);
    yolo_topk_nms<<<B_, 1024, smem, stream>>>(p, ancs, scores, out);
}